// GCN_82755429859753
// MI455X (gfx1250) — compile-verified
//
#include <hip/hip_runtime.h>

#define N_NODES 50000
#define N_EDGES 800000
#define DFEAT   128
#define ASTRIDE 132   // 16 rows * 132 floats: conflict-free LDS for f32-WMMA A fragments

typedef float v2f __attribute__((ext_vector_type(2)));
typedef float v8f __attribute__((ext_vector_type(8)));

// ---------------------------------------------------------------- utilities
__device__ __forceinline__ void atomic_add_f32(float* p, float v) {
    // native non-returning global_atomic_add_f32 (agent scope, relaxed)
    __hip_atomic_fetch_add(p, v, __ATOMIC_RELAXED, __HIP_MEMORY_SCOPE_AGENT);
}

__global__ __launch_bounds__(256) void zero_f4(float4* __restrict__ p, int n4) {
    int i = blockIdx.x * blockDim.x + threadIdx.x;
    if (i < n4) p[i] = make_float4(0.f, 0.f, 0.f, 0.f);
}

// ---------------------------------------------------------------- degrees
__global__ __launch_bounds__(256) void degree_kernel(const int* __restrict__ src,
                                                     const int* __restrict__ dst,
                                                     float* __restrict__ deg_out,
                                                     float* __restrict__ deg_in,
                                                     int n_edges) {
    int e = blockIdx.x * blockDim.x + threadIdx.x;
    if (e < n_edges) {
        atomic_add_f32(&deg_out[src[e]], 1.0f);
        atomic_add_f32(&deg_in [dst[e]], 1.0f);
    }
}

__global__ __launch_bounds__(256) void deg_to_rsqrt(float* __restrict__ p, int n) {
    int i = blockIdx.x * blockDim.x + threadIdx.x;
    if (i < n) p[i] = rsqrtf(fmaxf(p[i], 1.0f));
}

// ---------------------------------------------------------------- edge scatter
// one wave (32 lanes) per edge; lane handles 4 of the 128 features.
// Applies rs_out[src] on the fly so the normalized feature matrix is never
// materialized (saves a full N*D pass per layer).
__global__ __launch_bounds__(256) void scatter_kernel(const float* __restrict__ X,
                                                      const float* __restrict__ rs_out,
                                                      const int* __restrict__ src,
                                                      const int* __restrict__ dst,
                                                      float* __restrict__ msg,
                                                      int n_edges) {
    int e    = (blockIdx.x * blockDim.x + threadIdx.x) >> 5;
    int lane = threadIdx.x & 31;
    if (e >= n_edges) return;
    int   s = src[e];
    int   d = dst[e];
    float scale = rs_out[s];
    float4 v = ((const float4*)(X + (size_t)s * DFEAT))[lane];
    float* mp = msg + (size_t)d * DFEAT + lane * 4;
    atomic_add_f32(mp + 0, v.x * scale);
    atomic_add_f32(mp + 1, v.y * scale);
    atomic_add_f32(mp + 2, v.z * scale);
    atomic_add_f32(mp + 3, v.w * scale);
}

// ---------------------------------------------------------------- WMMA GEMM
// out[r,:] = act( rs_in[r] * (msg[r,:] @ W) + b )   using V_WMMA_F32_16X16X4_F32.
// Block: 256 threads = 8 waves; 16 node rows per block; wave w owns cols [16w,16w+16).
// rs_in is folded into the LDS staging of A (mathematically identical).
template <bool RELU>
__global__ __launch_bounds__(256) void gcn_gemm(const float* __restrict__ A,
                                                const float* __restrict__ rs_in,
                                                const float* __restrict__ W,
                                                const float* __restrict__ bias,
                                                float* __restrict__ out) {
    __shared__ float Alds[16 * ASTRIDE];

    const int tid     = threadIdx.x;
    const int wave    = tid >> 5;
    const int lane    = tid & 31;
    const int rowBase = blockIdx.x << 4;

    // stage A tile (16 x 128) into LDS, pre-scaled by rs_in[row]
    {
        const int r = tid >> 4;            // 0..15
        const int c = (tid & 15) << 3;     // 0,8,...,120
        const float scale = rs_in[rowBase + r];
        const float4* gp = (const float4*)(A + (size_t)(rowBase + r) * DFEAT + c);
        float4 a0 = gp[0];
        float4 a1 = gp[1];
        float* lp = &Alds[r * ASTRIDE + c];
        lp[0] = a0.x * scale; lp[1] = a0.y * scale; lp[2] = a0.z * scale; lp[3] = a0.w * scale;
        lp[4] = a1.x * scale; lp[5] = a1.y * scale; lp[6] = a1.z * scale; lp[7] = a1.w * scale;
    }
    __syncthreads();

    // f32 WMMA fragment addressing (ISA 7.12.2):
    //   A 16x4 : lane -> M = lane%16, K-pair base = 2*(lane/16)
    //   B 4x16 : lane -> N = lane%16, K-pair base = 2*(lane/16)
    const int   mrow = lane & 15;
    const int   khi  = (lane >> 4) << 1;       // 0 or 2
    const int   n0   = wave << 4;
    const float* wp  = W + n0 + mrow;          // + (k+khi)*128 per step
    const float* ap  = &Alds[mrow * ASTRIDE + khi];

    v8f acc = {0.f, 0.f, 0.f, 0.f, 0.f, 0.f, 0.f, 0.f};
#pragma unroll
    for (int k = 0; k < DFEAT; k += 4) {
        v2f a = *(const v2f*)(ap + k);         // ds_load_b64, bank-conflict free
        v2f b;
        b.x = wp[(size_t)(k + khi)     * DFEAT];
        b.y = wp[(size_t)(k + khi + 1) * DFEAT];
        acc = __builtin_amdgcn_wmma_f32_16x16x4_f32(false, a, false, b,
                                                    (short)0, acc, false, false);
    }

    // epilogue: C/D layout -> VGPR v holds row (v + 8*(lane/16)), col = n0 + lane%16
    const int   col  = n0 + mrow;
    const float bv   = bias[col];
    const int   rsel = (lane >> 4) << 3;
#pragma unroll
    for (int v = 0; v < 8; ++v) {
        float h = acc[v] + bv;
        if (RELU) h = fmaxf(h, 0.f);
        out[(size_t)(rowBase + v + rsel) * DFEAT + col] = h;
    }
}

// ---------------------------------------------------------------- launcher
extern "C" void kernel_launch(void* const* d_in, const int* in_sizes, int n_in,
                              void* d_out, int out_size, void* d_ws, size_t ws_size,
                              hipStream_t stream) {
    const float* x   = (const float*)d_in[0];
    const int*   src = (const int*)  d_in[1];
    const int*   dst = (const int*)  d_in[2];
    const float* W1  = (const float*)d_in[3];
    const float* b1  = (const float*)d_in[4];
    const float* W2  = (const float*)d_in[5];
    const float* b2  = (const float*)d_in[6];
    float*       out = (float*)d_out;

    // workspace layout (floats): deg/rs_out [0,N), deg/rs_in [N,2N),
    // msg [2N, 2N+N*D), h1 [2N+N*D, 2N+2*N*D)   => ~51.6 MB
    float* ws     = (float*)d_ws;
    float* rs_out = ws;
    float* rs_in  = ws + N_NODES;
    float* msg    = ws + 2 * N_NODES;
    float* h1     = msg + (size_t)N_NODES * DFEAT;

    const int T     = 256;
    const int ND4   = N_NODES * DFEAT / 4;       // float4 count for msg
    const int EBLK  = (N_EDGES + 7) / 8;         // 8 edges (waves) per block
    const int GEMMB = N_NODES / 16;              // 50000 = 3125 * 16 exactly

    // degrees -> rsqrt(clamp(deg,1))
    zero_f4<<<(2 * N_NODES / 4 + T - 1) / T, T, 0, stream>>>((float4*)rs_out, 2 * N_NODES / 4);
    degree_kernel<<<(N_EDGES + T - 1) / T, T, 0, stream>>>(src, dst, rs_out, rs_in, N_EDGES);
    deg_to_rsqrt<<<(2 * N_NODES + T - 1) / T, T, 0, stream>>>(rs_out, 2 * N_NODES);

    // layer 1: msg = segsum(x[src]*rs_out[src]); h1 = relu(rs_in*(msg@W1)+b1)
    zero_f4<<<(ND4 + T - 1) / T, T, 0, stream>>>((float4*)msg, ND4);
    scatter_kernel<<<EBLK, T, 0, stream>>>(x, rs_out, src, dst, msg, N_EDGES);
    gcn_gemm<true><<<GEMMB, T, 0, stream>>>(msg, rs_in, W1, b1, h1);

    // layer 2: msg = segsum(h1[src]*rs_out[src]); out = rs_in*(msg@W2)+b2
    zero_f4<<<(ND4 + T - 1) / T, T, 0, stream>>>((float4*)msg, ND4);
    scatter_kernel<<<EBLK, T, 0, stream>>>(h1, rs_out, src, dst, msg, N_EDGES);
    gcn_gemm<false><<<GEMMB, T, 0, stream>>>(msg, rs_in, W2, b2, out);
}